// GATClassifier_58780922413864
// MI455X (gfx1250) — compile-verified
//
#include <hip/hip_runtime.h>
#include <hip/hip_bf16.h>
#include <math.h>

typedef __attribute__((ext_vector_type(2))) float v2f;
typedef __attribute__((ext_vector_type(8))) float v8f;

#define NEG_SLOPE 0.2f
#define EPS_F 1e-16f

// ---------------------------------------------------------------------------
// Generic fp32 WMMA GEMM: C[M,N] = A[M,K] * B[K,N], all row-major.
// M % 16 == 0, N % 16 == 0, K % 4 == 0. One wave32 per 16x16 output tile.
// Uses V_WMMA_F32_16X16X4_F32 (full fp32 precision).
// ---------------------------------------------------------------------------
__global__ void gemm_f32_wmma(const float* __restrict__ A,
                              const float* __restrict__ B,
                              float* __restrict__ C,
                              int M, int N, int K) {
    const int wave = (int)((blockIdx.x * blockDim.x + threadIdx.x) >> 5);
    const int lane = (int)(threadIdx.x & 31);
    const int tilesN = N >> 4;
    const int tm = wave / tilesN;
    const int tn = wave % tilesN;
    if (tm >= (M >> 4)) return;                 // wave-uniform exit (EXEC stays full)

    const int row   = tm * 16 + (lane & 15);    // A row for this lane
    const int col   = tn * 16 + (lane & 15);    // B/C column for this lane
    const int khalf = (lane >> 4) * 2;          // lanes 0-15 -> K{0,1}; 16-31 -> K{2,3}

    const float* __restrict__ arow = A + (size_t)row * K;

    v8f acc = {};
    for (int k0 = 0; k0 < K; k0 += 4) {
        v2f a, b;
        a.x = arow[k0 + khalf];
        a.y = arow[k0 + khalf + 1];
        b.x = B[(size_t)(k0 + khalf) * N + col];
        b.y = B[(size_t)(k0 + khalf + 1) * N + col];
        acc = __builtin_amdgcn_wmma_f32_16x16x4_f32(
            /*neg_a=*/false, a, /*neg_b=*/false, b,
            /*c_mod=*/(short)0, acc, /*reuse_a=*/false, /*reuse_b=*/false);
    }

    // D layout: VGPR r -> lanes 0-15: M=r, lanes 16-31: M=r+8
    const int rbase = tm * 16 + ((lane >> 4) << 3);
#pragma unroll
    for (int r = 0; r < 8; ++r) {
        C[(size_t)(rbase + r) * N + col] = acc[r];
    }
}

// ---------------------------------------------------------------------------
// alpha_s[n,h] = sum_c h[n,h,c]*a_src[h,c];  alpha_d likewise.
// ---------------------------------------------------------------------------
__global__ void alpha_kernel(const float* __restrict__ h,
                             const float* __restrict__ a_src,
                             const float* __restrict__ a_dst,
                             float* __restrict__ out_s,
                             float* __restrict__ out_d,
                             int N, int H, int Cc) {
    int i = (int)(blockIdx.x * blockDim.x + threadIdx.x);
    if (i >= N * H) return;
    int n = i / H, hh = i % H;
    const float* hp = h + (size_t)n * H * Cc + (size_t)hh * Cc;
    const float* s  = a_src + (size_t)hh * Cc;
    const float* d  = a_dst + (size_t)hh * Cc;
    float ss = 0.f, dd = 0.f;
    for (int c = 0; c < Cc; ++c) {
        float v = hp[c];
        ss = fmaf(v, s[c], ss);
        dd = fmaf(v, d[c], dd);
    }
    out_s[i] = ss;
    out_d[i] = dd;
}

__global__ void fill_kernel(float* __restrict__ p, float v, size_t n) {
    size_t i = (size_t)blockIdx.x * blockDim.x + threadIdx.x;
    if (i < n) p[i] = v;
}

__device__ inline void atomicMaxF(float* addr, float val) {
    int old = __float_as_int(*(volatile float*)addr);
    while (__int_as_float(old) < val) {
        int assumed = old;
        old = atomicCAS((int*)addr, assumed, __float_as_int(val));
        if (old == assumed) break;
    }
}

__device__ inline void edge_endpoints(const int* __restrict__ ei, int E, int e,
                                      int& s, int& d) {
    if (e < E) { s = ei[e]; d = ei[E + e]; }
    else       { s = e - E; d = e - E; }        // self-loops appended
}

// Pass 1: segment max of leaky_relu(alpha_s[src] + alpha_d[dst]) per dst,head
__global__ void edge_max_kernel(const int* __restrict__ ei, int E, int Etot,
                                const float* __restrict__ as,
                                const float* __restrict__ ad,
                                float* __restrict__ maxb, int H) {
    int i = (int)(blockIdx.x * blockDim.x + threadIdx.x);
    if (i >= Etot * H) return;
    int e = i / H, hh = i % H;
    int s, d; edge_endpoints(ei, E, e, s, d);
    float el = as[(size_t)s * H + hh] + ad[(size_t)d * H + hh];
    el = el >= 0.f ? el : NEG_SLOPE * el;
    atomicMaxF(&maxb[(size_t)d * H + hh], el);
}

// Pass 2: ex = exp(e - max); store ex, segment-sum into denom
__global__ void edge_sum_kernel(const int* __restrict__ ei, int E, int Etot,
                                const float* __restrict__ as,
                                const float* __restrict__ ad,
                                const float* __restrict__ maxb,
                                float* __restrict__ denom,
                                float* __restrict__ exbuf, int H) {
    int i = (int)(blockIdx.x * blockDim.x + threadIdx.x);
    if (i >= Etot * H) return;
    int e = i / H, hh = i % H;
    int s, d; edge_endpoints(ei, E, e, s, d);
    float el = as[(size_t)s * H + hh] + ad[(size_t)d * H + hh];
    el = el >= 0.f ? el : NEG_SLOPE * el;
    float ex = expf(el - maxb[(size_t)d * H + hh]);
    exbuf[i] = ex;
    atomicAdd(&denom[(size_t)d * H + hh], ex);
}

// Pass 3: out[dst] += alpha * h[src].  One wave per edge; lanes stride the
// H*C feature vector -> coalesced 128B gathers + coalesced f32 atomics (L2).
__global__ void edge_aggr_kernel(const int* __restrict__ ei, int E, int Etot,
                                 const float* __restrict__ hfeat,
                                 const float* __restrict__ exbuf,
                                 const float* __restrict__ denom,
                                 float* __restrict__ acc, int H, int Cc) {
    int wave = (int)((blockIdx.x * blockDim.x + threadIdx.x) >> 5);
    int lane = (int)(threadIdx.x & 31);
    if (wave >= Etot) return;                   // wave-uniform
    int s, d; edge_endpoints(ei, E, wave, s, d);
    const int F = H * Cc;
    for (int j = lane; j < F; j += 32) {
        int hh = j / Cc;
        float alpha = exbuf[(size_t)wave * H + hh] /
                      (denom[(size_t)d * H + hh] + EPS_F);
        atomicAdd(&acc[(size_t)d * F + j], alpha * hfeat[(size_t)s * F + j]);
    }
}

// act = elu(acc + bias) in place
__global__ void bias_elu_kernel(float* __restrict__ a,
                                const float* __restrict__ b,
                                size_t n, int F) {
    size_t i = (size_t)blockIdx.x * blockDim.x + threadIdx.x;
    if (i >= n) return;
    float v = a[i] + b[i % F];
    a[i] = v > 0.f ? v : expm1f(v);
}

// out = acc + bias
__global__ void bias_out_kernel(const float* __restrict__ a,
                                const float* __restrict__ b,
                                float* __restrict__ out,
                                size_t n, int F) {
    size_t i = (size_t)blockIdx.x * blockDim.x + threadIdx.x;
    if (i >= n) return;
    out[i] = a[i] + b[i % F];
}

static inline int cdiv(long long a, int b) { return (int)((a + b - 1) / b); }

extern "C" void kernel_launch(void* const* d_in, const int* in_sizes, int n_in,
                              void* d_out, int out_size, void* d_ws, size_t ws_size,
                              hipStream_t stream) {
    const float* x        = (const float*)d_in[0];
    const int*   ei       = (const int*)d_in[1];   // [2,E] int32 (JAX x64 off)
    const float* W1       = (const float*)d_in[2];
    const float* att_s1   = (const float*)d_in[3];
    const float* att_d1   = (const float*)d_in[4];
    const float* b1       = (const float*)d_in[5];
    const float* W2       = (const float*)d_in[6];
    const float* att_s2   = (const float*)d_in[7];
    const float* att_d2   = (const float*)d_in[8];
    const float* b2       = (const float*)d_in[9];
    float*       out      = (float*)d_out;

    const int IN = 256, H1 = 8, C1 = 32, F1 = H1 * C1;   // 256
    const int H2 = 1, C2 = 16, F2 = H2 * C2;             // 16
    const int N    = in_sizes[0] / IN;                   // 50000
    const int E    = in_sizes[1] / 2;                    // 800000
    const int Etot = E + N;                              // + self loops

    // ---- workspace carve ----
    float* w = (float*)d_ws;
    float* h1   = w; w += (size_t)N * F1;
    float* acc1 = w; w += (size_t)N * F1;
    float* as1  = w; w += (size_t)N * H1;
    float* ad1  = w; w += (size_t)N * H1;
    float* mx1  = w; w += (size_t)N * H1;
    float* dn1  = w; w += (size_t)N * H1;
    float* ex1  = w; w += (size_t)Etot * H1;
    float* h2   = w; w += (size_t)N * F2;
    float* as2  = w; w += (size_t)N;
    float* ad2  = w; w += (size_t)N;
    float* mx2  = w; w += (size_t)N;
    float* dn2  = w; w += (size_t)N;
    float* ex2  = w; w += (size_t)Etot;
    float* acc2 = w; w += (size_t)N * F2;

    const int B = 256;

    // =================== layer 1 (heads=8, C=32, concat) ===================
    {   // GEMM1: h1 = x @ W1   [N,256]x[256,256]
        int tiles = (N / 16) * (F1 / 16);
        gemm_f32_wmma<<<cdiv(tiles, 8), B, 0, stream>>>(x, W1, h1, N, F1, IN);
    }
    alpha_kernel<<<cdiv((long long)N * H1, B), B, 0, stream>>>(h1, att_s1, att_d1,
                                                               as1, ad1, N, H1, C1);
    fill_kernel<<<cdiv((long long)N * H1, B), B, 0, stream>>>(mx1, -INFINITY, (size_t)N * H1);
    fill_kernel<<<cdiv((long long)N * H1, B), B, 0, stream>>>(dn1, 0.f, (size_t)N * H1);
    fill_kernel<<<cdiv((long long)N * F1, B), B, 0, stream>>>(acc1, 0.f, (size_t)N * F1);

    edge_max_kernel<<<cdiv((long long)Etot * H1, B), B, 0, stream>>>(ei, E, Etot,
                                                                     as1, ad1, mx1, H1);
    edge_sum_kernel<<<cdiv((long long)Etot * H1, B), B, 0, stream>>>(ei, E, Etot,
                                                                     as1, ad1, mx1, dn1, ex1, H1);
    edge_aggr_kernel<<<cdiv((long long)Etot * 32, B), B, 0, stream>>>(ei, E, Etot,
                                                                      h1, ex1, dn1, acc1, H1, C1);
    bias_elu_kernel<<<cdiv((long long)N * F1, B), B, 0, stream>>>(acc1, b1, (size_t)N * F1, F1);

    // =================== layer 2 (heads=1, C=16, mean==identity) ===========
    {   // GEMM2: h2 = act1 @ W2  [N,256]x[256,16]
        int tiles = (N / 16) * (F2 / 16);
        gemm_f32_wmma<<<cdiv(tiles, 8), B, 0, stream>>>(acc1, W2, h2, N, F2, IN);
    }
    alpha_kernel<<<cdiv((long long)N * H2, B), B, 0, stream>>>(h2, att_s2, att_d2,
                                                               as2, ad2, N, H2, C2);
    fill_kernel<<<cdiv((long long)N, B), B, 0, stream>>>(mx2, -INFINITY, (size_t)N);
    fill_kernel<<<cdiv((long long)N, B), B, 0, stream>>>(dn2, 0.f, (size_t)N);
    fill_kernel<<<cdiv((long long)N * F2, B), B, 0, stream>>>(acc2, 0.f, (size_t)N * F2);

    edge_max_kernel<<<cdiv((long long)Etot * H2, B), B, 0, stream>>>(ei, E, Etot,
                                                                     as2, ad2, mx2, H2);
    edge_sum_kernel<<<cdiv((long long)Etot * H2, B), B, 0, stream>>>(ei, E, Etot,
                                                                     as2, ad2, mx2, dn2, ex2, H2);
    edge_aggr_kernel<<<cdiv((long long)Etot * 32, B), B, 0, stream>>>(ei, E, Etot,
                                                                      h2, ex2, dn2, acc2, H2, C2);
    bias_out_kernel<<<cdiv((long long)N * F2, B), B, 0, stream>>>(acc2, b2, out,
                                                                  (size_t)N * F2, F2);
}